// RegressLoss_26774826123527
// MI455X (gfx1250) — compile-verified
//
#include <hip/hip_runtime.h>
#include <hip/hip_bf16.h>

#define PI_F 3.14159265358979323846f
#define BETA_F (1.0f / 9.0f)

// ---------------------------------------------------------------------------
// Sutherland–Hodgman: clip convex quad `a` against convex quad (px[4],py[4]),
// return intersection area (shoelace). Matches reference _clip_step semantics:
// inside = cross(d, p-e1) >= 0, t = s_prev / (|den|<1e-12 ? 1e-12 : den).
// ---------------------------------------------------------------------------
__device__ inline float clip_area(const float2 a[4], const float* px, const float* py) {
    float2 bufA[10], bufB[10];
    float2* cur = bufA;
    float2* nxt = bufB;
    int n = 4;
#pragma unroll
    for (int k = 0; k < 4; ++k) cur[k] = a[k];
#pragma unroll
    for (int e = 0; e < 4; ++e) {
        const float e1x = px[e], e1y = py[e];
        const float dx = px[(e + 1) & 3] - e1x;
        const float dy = py[(e + 1) & 3] - e1y;
        int m = 0;
        float2 prev = cur[n - 1];
        float sp = dx * (prev.y - e1y) - dy * (prev.x - e1x);
        for (int i = 0; i < n; ++i) {
            const float2 p = cur[i];
            const float sc = dx * (p.y - e1y) - dy * (p.x - e1x);
            const bool cin = (sc >= 0.0f);
            const bool pin = (sp >= 0.0f);
            if (cin != pin) {
                const float den = sp - sc;
                const float dd = (fabsf(den) < 1e-12f) ? 1e-12f : den;
                const float t = sp / dd;
                float2 ip;
                ip.x = prev.x + t * (p.x - prev.x);
                ip.y = prev.y + t * (p.y - prev.y);
                nxt[m++] = ip;
            }
            if (cin) nxt[m++] = p;
            prev = p;
            sp = sc;
        }
        float2* tswap = cur; cur = nxt; nxt = tswap;
        n = m;
        if (n == 0) return 0.0f;
    }
    float area = 0.0f;
    const float2 p0 = cur[0];
    float2 pp = p0;
    for (int i = 1; i < n; ++i) {
        const float2 q = cur[i];
        area += pp.x * q.y - q.x * pp.y;
        pp = q;
    }
    area += pp.x * p0.y - p0.x * pp.y;
    return 0.5f * fabsf(area);
}

// ---------------------------------------------------------------------------
// Kernel 1: per-gt derived data (square, area, tan, center/size, corners) and
// zero-init of packed column maxima.  gtd layout (20 floats per gt):
// [0..3] min-area square, [4] w*h, [5] tan(theta), [6..9] cx,cy,w,h,
// [10..13] corner x, [14..17] corner y, [18..19] pad.
// ---------------------------------------------------------------------------
__global__ void rr_prep_kernel(const float* __restrict__ ann,
                               float* __restrict__ gtd,
                               unsigned long long* __restrict__ colmax,
                               int BM) {
    const int t = threadIdx.x;
    if (t >= BM) return;
    const float* g = ann + (size_t)t * 6;
    const float cx = g[0], cy = g[1], w = g[2], h = g[3], th = g[4];
    float* o = gtd + (size_t)t * 20;
    const float s = fmaxf(w, h) * 0.5f;
    o[0] = cx - s; o[1] = cy - s; o[2] = cx + s; o[3] = cy + s;
    o[4] = w * h;
    const float a = th * (PI_F / 180.0f);
    o[5] = tanf(a);
    o[6] = cx; o[7] = cy; o[8] = w; o[9] = h;
    float sn, cs;
    sincosf(a, &sn, &cs);
    const float hdx = 0.5f * w, hdy = 0.5f * h;
    const float lx[4] = {-hdx, hdx, hdx, -hdx};
    const float ly[4] = {-hdy, -hdy, hdy, hdy};
#pragma unroll
    for (int k = 0; k < 4; ++k) {
        o[10 + k] = cx + cs * lx[k] - sn * ly[k];
        o[14 + k] = cy + sn * lx[k] + cs * ly[k];
    }
    o[18] = 0.0f; o[19] = 0.0f;
    colmax[t] = 0ull;
}

// ---------------------------------------------------------------------------
// Kernel 2: one thread per anchor. Stage gt data to LDS with CDNA5 async
// global->LDS B128 copies (ASYNCcnt path), then gated rotated-IoU over all
// gts, packed-u64 column max via LDS ds_max_u64, row max/argmax, smooth-L1.
// ---------------------------------------------------------------------------
__global__ __launch_bounds__(256) void rr_pair_kernel(
    const float* __restrict__ reg,
    const float* __restrict__ anch,
    const float* __restrict__ gtd,
    unsigned long long* __restrict__ colmax,
    float2* __restrict__ rowdata,
    int N, int M) {
    __shared__ __align__(16) float gl[32 * 20];
    __shared__ unsigned long long cmax[32];

    const int tid = threadIdx.x;
    const int blocksPerImage = N >> 8;
    const int b = blockIdx.x / blocksPerImage;
    const int i = (blockIdx.x % blocksPerImage) * 256 + tid;
    const int nChunks = (M * 20) / 4;  // 16-byte chunks (M*20 floats, M mult of 4 here)
    const float* src = gtd + (size_t)b * (M * 20);

    // CDNA5 async global->LDS broadcast of the per-gt table (2.5KB, B128/lane).
    for (int k = tid; k < nChunks; k += 256) {
        unsigned int laddr = (unsigned int)(size_t)&gl[k * 4];
        unsigned long long gaddr = (unsigned long long)(size_t)(src + k * 4);
        asm volatile("global_load_async_to_lds_b128 %0, %1, off"
                     :: "v"(laddr), "v"(gaddr) : "memory");
    }
    asm volatile("s_wait_asynccnt 0" ::: "memory");
    if (tid < M) cmax[tid] = 0ull;

    // Prefetch the regression row now; it is consumed only after the gt loop.
    const float* rp = reg + (size_t)(b * N + i) * 5;
    __builtin_prefetch(rp, 0, 0);

    __syncthreads();

    // Per-anchor derived data.
    const float* ap = anch + (size_t)(b * N + i) * 5;
    const float acx = ap[0], acy = ap[1], aw = ap[2], ah = ap[3], ath = ap[4];
    const float asz = fmaxf(aw, ah) * 0.5f;
    const float ax0 = acx - asz, ay0 = acy - asz;
    const float ax1 = acx + asz, ay1 = acy + asz;
    const float sqAreaA = (ax1 - ax0) * (ay1 - ay0);
    const float areaA = aw * ah;
    const float arad = ath * (PI_F / 180.0f);
    float sn, cs;
    sincosf(arad, &sn, &cs);
    const float atn = tanf(arad);
    const float hdx = 0.5f * aw, hdy = 0.5f * ah;
    const float lx[4] = {-hdx, hdx, hdx, -hdx};
    const float ly[4] = {-hdy, -hdy, hdy, hdy};
    float2 pa[4];
#pragma unroll
    for (int k = 0; k < 4; ++k) {
        pa[k].x = acx + cs * lx[k] - sn * ly[k];
        pa[k].y = acy + sn * lx[k] + cs * ly[k];
    }

    float best = -1.0f;
    int bestj = 0;
    for (int j = 0; j < M; ++j) {
        const float* g = &gl[j * 20];
        // indicator: AABB IoU of min-area squares
        const float ltx = fmaxf(ax0, g[0]), lty = fmaxf(ay0, g[1]);
        const float rbx = fminf(ax1, g[2]), rby = fminf(ay1, g[3]);
        const float wx = fmaxf(rbx - ltx, 0.0f), wy = fmaxf(rby - lty, 0.0f);
        const float inter = wx * wy;
        const float sqAreaB = (g[2] - g[0]) * (g[3] - g[1]);
        const float ind = inter / (sqAreaA + sqAreaB - inter + 1e-9f);
        float ov = 0.0f;
        if (ind >= 0.1f) {
            const float ia = clip_area(pa, g + 10, g + 14);
            ov = ia / (areaA + g[4] - ia + 1e-9f);
        }
        // column max with first-index tie-break (ov >= 0 so f32 bits order as u32)
        const unsigned long long pk =
            ((unsigned long long)__float_as_uint(ov) << 32) |
            (unsigned long long)(0xFFFFFFFFu - (unsigned int)i);
        atomicMax(&cmax[j], pk);
        if (ov > best) { best = ov; bestj = j; }
    }

    // encode(anchor, gt[bestj]) + smooth-L1 sum over 5 coords
    const float* g = &gl[bestj * 20];
    float tt[5];
    tt[0] = 10.0f * (g[6] - acx) / aw;
    tt[1] = 10.0f * (g[7] - acy) / ah;
    tt[2] = 5.0f * logf(g[8] / aw);
    tt[3] = 5.0f * logf(g[9] / ah);
    tt[4] = 15.0f * (g[5] - atn);
    float sl = 0.0f;
#pragma unroll
    for (int k = 0; k < 5; ++k) {
        const float d = fabsf(rp[k] - tt[k]);
        sl += (d < BETA_F) ? (0.5f * d * d / BETA_F) : (d - 0.5f * BETA_F);
    }
    rowdata[(size_t)b * N + i] = make_float2(best, sl);

    __syncthreads();
    if (tid < M) atomicMax(&colmax[b * M + tid], cmax[tid]);
}

// ---------------------------------------------------------------------------
// Kernel 3: per-image reduction. Decode forced anchors (column max < 0.5),
// count positives, deterministic LDS tree reduce of the masked loss.
// ---------------------------------------------------------------------------
__global__ __launch_bounds__(1024) void rr_reduce_kernel(
    const float2* __restrict__ rowdata,
    const unsigned long long* __restrict__ colmax,
    float* __restrict__ lossb,
    int N, int M) {
    const int b = blockIdx.x;
    const int tid = threadIdx.x;
    __shared__ unsigned int forced_s[32];
    __shared__ float ssum[1024];
    __shared__ int scnt[1024];

    if (tid < 32) {
        unsigned int f = 0xFFFFFFFFu;
        if (tid < M) {
            const unsigned long long pk = colmax[b * M + tid];
            const float v = __uint_as_float((unsigned int)(pk >> 32));
            const unsigned int idx = 0xFFFFFFFFu - (unsigned int)(pk & 0xFFFFFFFFull);
            if (v < 0.5f) f = idx;
        }
        forced_s[tid] = f;
    }
    __syncthreads();

    unsigned int forced[32];
#pragma unroll
    for (int j = 0; j < 32; ++j) forced[j] = forced_s[j];

    float sum = 0.0f;
    int cnt = 0;
    for (int i = tid; i < N; i += 1024) {
        const float2 rd = rowdata[(size_t)b * N + i];
        bool pos = rd.x >= 0.5f;
#pragma unroll
        for (int j = 0; j < 32; ++j) pos = pos || (forced[j] == (unsigned int)i);
        if (pos) { cnt += 1; sum += rd.y; }
    }
    ssum[tid] = sum;
    scnt[tid] = cnt;
    __syncthreads();
    for (int off = 512; off > 0; off >>= 1) {
        if (tid < off) {
            ssum[tid] += ssum[tid + off];
            scnt[tid] += scnt[tid + off];
        }
        __syncthreads();
    }
    if (tid == 0) {
        const int np = scnt[0];
        const int denom = (np * 5 > 1) ? np * 5 : 1;
        lossb[b] = (np > 0) ? (ssum[0] / (float)denom) : 0.0f;
    }
}

__global__ void rr_final_kernel(const float* __restrict__ lossb,
                                float* __restrict__ out, int B) {
    if (threadIdx.x == 0 && blockIdx.x == 0) {
        float s = 0.0f;
        for (int b = 0; b < B; ++b) s += lossb[b];
        out[0] = s / (float)B;
    }
}

// ---------------------------------------------------------------------------
extern "C" void kernel_launch(void* const* d_in, const int* in_sizes, int n_in,
                              void* d_out, int out_size, void* d_ws, size_t ws_size,
                              hipStream_t stream) {
    const float* reg  = (const float*)d_in[0];  // [B,N,5]
    const float* anch = (const float*)d_in[1];  // [B,N,5]
    const float* ann  = (const float*)d_in[2];  // [B,M,6]
    float* out = (float*)d_out;

    const int B = 2;
    const int N = in_sizes[1] / (B * 5);
    const int M = in_sizes[2] / (B * 6);

    char* ws = (char*)d_ws;
    size_t off = 0;
    float* gtd = (float*)(ws + off);                 off += (size_t)B * M * 20 * sizeof(float);
    off = (off + 15) & ~(size_t)15;
    unsigned long long* colmax = (unsigned long long*)(ws + off);
    off += (size_t)B * M * sizeof(unsigned long long);
    off = (off + 15) & ~(size_t)15;
    float2* rowdata = (float2*)(ws + off);           off += (size_t)B * N * sizeof(float2);
    float* lossb = (float*)(ws + off);               off += (size_t)B * sizeof(float);

    rr_prep_kernel<<<1, B * M, 0, stream>>>(ann, gtd, colmax, B * M);

    const int blocksPerImage = N / 256;
    rr_pair_kernel<<<B * blocksPerImage, 256, 0, stream>>>(
        reg, anch, gtd, colmax, rowdata, N, M);

    rr_reduce_kernel<<<B, 1024, 0, stream>>>(rowdata, colmax, lossb, N, M);

    rr_final_kernel<<<1, 1, 0, stream>>>(lossb, out, B);
}